// NoRollCaTentLayer_66374424593117
// MI455X (gfx1250) — compile-verified
//
#include <hip/hip_runtime.h>
#include <math.h>

// CDNA5 / gfx1250, wave32.
typedef float v2f __attribute__((ext_vector_type(2)));
typedef float v8f __attribute__((ext_vector_type(8)));

#define TILE_T 256                 // outputs per block in time = 16 (M) x 16 (N)
#define TILE_C 16                  // channels per block
#define FW     64                  // filter taps
#define PADLO  31                  // SAME left pad = (FW-1)/2
#define NQ     20                  // K chunks of 4 -> k = 0..79 covers m(15)+j(63)=78
#define XROWS  (240 + 4 * NQ)      // staged input rows: max lt = 16*15 + 4*NQ-1

// Guard-banded Toeplitz filter: index = (j + JPAD), j in [-15, 80)
#define JPAD   15
#define FROWS  96                  // 15 zero rows + 64 taps + 17 zero rows

// Depthwise temporal conv via banded-Toeplitz WMMA:
//   A[m,k] = filt[k-m] (per channel), B[k,n] = x[t0+16n+k-31] (per channel)
//   D[m,n] = pre[t0+16n+m]
__global__ __launch_bounds__(256, 2)
void catent_depthwise_wmma(const float* __restrict__ x,
                           const float* __restrict__ w,
                           const float* __restrict__ bias,
                           float* __restrict__ out,
                           int T, int C)
{
    __shared__ float xs[XROWS * 16];      // input tile, xor-swizzled columns
    __shared__ float flt2[FROWS * 17];    // guard-banded filter, 17-pad (17 coprime 64)
    __shared__ float outs[TILE_T * 16];   // output staging, xor-swizzled columns
    __shared__ float partial[16 * 16];    // [group][channel] sumsq partials
    __shared__ float biasv[16];
    __shared__ float rnorm[16];

    const int tid = threadIdx.x;
    const int c0  = blockIdx.x * TILE_C;
    const int t0  = blockIdx.y * TILE_T;

    // ---- stage x[t0-31 .. t0+XROWS-32) x [c0..c0+16) into LDS (b128 global loads) ----
    {
        const int lt_base = tid >> 2;     // 0..63 rows per round
        const int quad    = tid & 3;      // 4 threads per 64B row
        for (int base = 0; base < XROWS; base += 64) {
            const int lt = base + lt_base;
            const int gt = t0 - PADLO + lt;
            float4 v = {0.f, 0.f, 0.f, 0.f};
            if (gt >= 0 && gt < T)
                v = *reinterpret_cast<const float4*>(x + (size_t)gt * C + c0 + quad * 4);
            const int sw = (lt >> 4) & 15;      // xor swizzle -> conflict-free B reads
            float* row = &xs[lt * 16];
            row[(quad * 4 + 0) ^ sw] = v.x;
            row[(quad * 4 + 1) ^ sw] = v.y;
            row[(quad * 4 + 2) ^ sw] = v.z;
            row[(quad * 4 + 3) ^ sw] = v.w;
        }
    }

    // ---- zero the guard bands (rows 0..14 and 79..95), load raw taps into 15..78 ----
    for (int i = tid; i < 32 * 17; i += 256) {
        const int r = i / 17, c = i % 17;
        const int row = (r < JPAD) ? r : (r + FW);    // 0..14 and 79..95
        flt2[row * 17 + c] = 0.f;
    }
    for (int idx = tid; idx < FW * 16; idx += 256) {
        const int f = idx >> 4, c = idx & 15;
        flt2[(f + JPAD) * 17 + c] = w[(size_t)f * C + c0 + c];
    }
    if (tid < 16) biasv[tid] = bias[c0 + tid];
    __syncthreads();

    // ---- weight-norm: two-stage parallel sum of squares over 64 taps ----
    {
        const int c = tid & 15, g = tid >> 4;         // 16 groups x 4 taps
        float s = 0.f;
#pragma unroll
        for (int ff = 0; ff < 4; ++ff) {
            const float v = flt2[(g * 4 + ff + JPAD) * 17 + c];
            s += v * v;
        }
        partial[g * 16 + c] = s;
    }
    __syncthreads();
    if (tid < 16) {
        float s = 0.f;
#pragma unroll
        for (int g = 0; g < 16; ++g) s += partial[g * 16 + tid];
        rnorm[tid] = 1.0f / fmaxf(sqrtf(s), 1e-8f);
    }
    __syncthreads();
    // ---- normalize + positive clamp (ReLU on the filter) ----
    for (int idx = tid; idx < FW * 16; idx += 256) {
        const int f = idx >> 4, c = idx & 15;
        flt2[(f + JPAD) * 17 + c] = fmaxf(0.f, flt2[(f + JPAD) * 17 + c] * rnorm[c]);
    }
    __syncthreads();

    // ---- WMMA compute: 8 waves x 2 channels each x 20 chained K=4 fp32 WMMAs ----
    const int lane = tid & 31;
    const int wv   = tid >> 5;     // wave id 0..7
    const int lo   = lane & 15;    // A row m / B col n / D col N
    const int hi   = lane >> 4;    // selects K pair {2hi, 2hi+1} (A,B) and M half (D)

    for (int cc = 0; cc < 2; ++cc) {
        const int cL = wv * 2 + cc;            // local channel 0..15
        const float bv = biasv[cL];            // fold bias into accumulator init
        v8f acc = {bv, bv, bv, bv, bv, bv, bv, bv};
        // 5 groups of 4 K-chunks: swizzle term constant per group, so all inner
        // addresses are base + immediate offset (A: +272B, B: +256B per step).
#pragma unroll
        for (int g = 0; g < 5; ++g) {
            const int swc   = cL ^ ((lo + g) & 15);
            const int bbase = 256 * lo + 32 * hi + 256 * g + swc;            // float idx
            const int abase = (16 * g + 2 * hi - lo + JPAD) * 17 + cL;       // float idx
#pragma unroll
            for (int qq = 0; qq < 4; ++qq) {
                const float ax = flt2[abase + qq * 68];        // taps j, j+1 (2addr pair)
                const float ay = flt2[abase + qq * 68 + 17];
                const float bx = xs[bbase + qq * 64];          // times k, k+1 (2addr pair)
                const float by = xs[bbase + qq * 64 + 16];
                const v2f av  = {ax, ay};
                const v2f bv2 = {bx, by};
                // v_wmma_f32_16x16x4_f32: full fp32 precision on the matrix pipe
                acc = __builtin_amdgcn_wmma_f32_16x16x4_f32(
                    false, av, false, bv2, (short)0, acc, false, false);
            }
        }
        // D layout: lane holds col N=lo, rows M = r + 8*hi  ->  time row = 16*lo + M
        // outs swizzle col = cL ^ ((row>>4)&15) = cL ^ lo  -> conflict-free writes
#pragma unroll
        for (int r = 0; r < 8; ++r) {
            const int row = 16 * lo + 8 * hi + r;
            outs[row * 16 + (cL ^ lo)] = acc[r];
        }
    }
    __syncthreads();

    // ---- coalesced 64B-per-row stores ----
    for (int idx = tid; idx < TILE_T * 4; idx += 256) {
        const int row = idx >> 2, quad = idx & 3;
        const int sw = (row >> 4) & 15;
        float4 v;
        v.x = outs[row * 16 + ((quad * 4 + 0) ^ sw)];
        v.y = outs[row * 16 + ((quad * 4 + 1) ^ sw)];
        v.z = outs[row * 16 + ((quad * 4 + 2) ^ sw)];
        v.w = outs[row * 16 + ((quad * 4 + 3) ^ sw)];
        *reinterpret_cast<float4*>(out + (size_t)(t0 + row) * C + c0 + quad * 4) = v;
    }
}

extern "C" void kernel_launch(void* const* d_in, const int* in_sizes, int n_in,
                              void* d_out, int out_size, void* d_ws, size_t ws_size,
                              hipStream_t stream) {
    const float* x = (const float*)d_in[0];   // [T, C] fp32
    const float* w = (const float*)d_in[1];   // [FW, C] fp32
    const float* b = (const float*)d_in[2];   // [C] fp32
    float* out     = (float*)d_out;           // [T, C] fp32

    const int C = in_sizes[2];
    const int T = in_sizes[0] / C;

    dim3 grid(C / TILE_C, T / TILE_T);        // 256 x 16 = 4096 blocks
    catent_depthwise_wmma<<<grid, 256, 0, stream>>>(x, w, b, out, T, C);
}